// Attribute_87926570484230
// MI455X (gfx1250) — compile-verified
//
#include <hip/hip_runtime.h>
#include <hip/hip_bf16.h>

typedef __attribute__((ext_vector_type(2))) float v2f;
typedef __attribute__((ext_vector_type(8))) float v8f;

#define B_ 64
#define L_ 1024
#define D_ 1024
#define S_ 32

// ---------------------------------------------------------------------------
// Kernel 1: segment sums via V_WMMA_F32_16X16X4_F32
//   seg_sum[b, s, d] = sum_l (attributes[b,l] == s+1) * feats[b, l, d]
// One workgroup = (batch b, 64-wide D slab). 4 waves, each owns 16 D columns
// and accumulates BOTH 16-row s-tiles (S = 32), so features stream once.
//   A (16x4 f32, 2 VGPRs): lane m = lane&15, VGPR k-pair = (lane<16)?{0,1}:{2,3}
//   B (4x16 f32, 2 VGPRs): mirrored k-pair layout, N = lane&15
//   C/D (16x16 f32, 8 VGPRs): row M = r + 8*(lane>=16), col N = lane&15
// ---------------------------------------------------------------------------
__global__ __launch_bounds__(128) void seg_sum_wmma_kernel(
    const int* __restrict__ attrs, const float* __restrict__ feats,
    float* __restrict__ seg_sum)
{
    __shared__ int s_attr[L_];

    const int b    = blockIdx.x >> 4;   // 16 slabs of 64 columns per batch
    const int slab = blockIdx.x & 15;
    const int tid  = threadIdx.x;
    const int wave = tid >> 5;
    const int lane = tid & 31;

    // stage attribute ids for this batch into LDS (4 KB)
    const int* ab = attrs + (size_t)b * L_;
    for (int i = tid; i < L_; i += 128) s_attr[i] = ab[i];
    __syncthreads();

    const int m     = lane & 15;            // A row / B,C column within tile
    const int koff  = (lane >> 4) << 1;     // 0 or 2: which K pair this half-wave holds
    const int d_col = slab * 64 + wave * 16 + m;
    const float* fb = feats + (size_t)b * L_ * D_ + d_col;

    v8f c0 = {};  // s-tile 0 (labels 1..16)
    v8f c1 = {};  // s-tile 1 (labels 17..32)

    for (int l0 = 0; l0 < L_; l0 += 4) {
        const int ax = s_attr[l0 + koff];
        const int ay = s_attr[l0 + koff + 1];

        v2f A0, A1, Bm;
        A0.x = (ax == m + 1)  ? 1.0f : 0.0f;
        A0.y = (ay == m + 1)  ? 1.0f : 0.0f;
        A1.x = (ax == m + 17) ? 1.0f : 0.0f;
        A1.y = (ay == m + 17) ? 1.0f : 0.0f;

        const float* fp = fb + (size_t)l0 * D_;
        Bm.x = fp[(size_t)koff * D_];
        Bm.y = fp[(size_t)(koff + 1) * D_];

        if (((l0 & 15) == 0) && (l0 + 16 < L_)) {
            __builtin_prefetch(fp + (size_t)16 * D_, 0, 0); // global_prefetch_b8
        }

        // 8-arg form: (neg_a, A, neg_b, B, c_mod, C, reuse_a, reuse_b)
        c0 = __builtin_amdgcn_wmma_f32_16x16x4_f32(false, A0, false, Bm,
                                                   (short)0, c0, false, false);
        c1 = __builtin_amdgcn_wmma_f32_16x16x4_f32(false, A1, false, Bm,
                                                   (short)0, c1, false, false);
    }

    const int mrow = (lane >> 4) << 3;  // 0 or 8
#pragma unroll
    for (int r = 0; r < 8; ++r) {
        const int s0 = mrow + r;
        seg_sum[((size_t)b * S_ + s0)      * D_ + d_col] = c0[r];
        seg_sum[((size_t)b * S_ + s0 + 16) * D_ + d_col] = c1[r];
    }
}

// ---------------------------------------------------------------------------
// Kernel 2: per-(b,s) count + fused dot/norm reduction -> cosine
//   cos = (dot(V, seg_sum)/cnt) / max(||V|| * sqrt(||seg_sum||^2)/cnt, 1e-8)
// ---------------------------------------------------------------------------
__global__ __launch_bounds__(256) void cos_kernel(
    const int* __restrict__ attrs, const float* __restrict__ Vgs,
    const float* __restrict__ seg_sum, float* __restrict__ cos_out)
{
    const int bs  = blockIdx.x;
    const int b   = bs >> 5;
    const int s   = bs & 31;
    const int tid = threadIdx.x;

    float cnt = 0.0f, num = 0.0f, nv = 0.0f, nm = 0.0f;

    const int* ab = attrs + (size_t)b * L_;
    for (int l = tid; l < L_; l += 256) cnt += (ab[l] == s + 1) ? 1.0f : 0.0f;

    const float* v  = Vgs     + (size_t)bs * D_;
    const float* ms = seg_sum + (size_t)bs * D_;
    for (int d = tid; d < D_; d += 256) {
        const float vv = v[d], mm = ms[d];
        num += vv * mm;
        nv  += vv * vv;
        nm  += mm * mm;
    }

    __shared__ float r0[256], r1[256], r2[256], r3[256];
    r0[tid] = num; r1[tid] = nv; r2[tid] = nm; r3[tid] = cnt;
    __syncthreads();
    for (int off = 128; off > 0; off >>= 1) {
        if (tid < off) {
            r0[tid] += r0[tid + off];
            r1[tid] += r1[tid + off];
            r2[tid] += r2[tid + off];
            r3[tid] += r3[tid + off];
        }
        __syncthreads();
    }
    if (tid == 0) {
        const float c        = r3[0];
        const float dot_mean = r0[0] / c;
        const float denom    = fmaxf(sqrtf(r1[0]) * sqrtf(r2[0]) / c, 1e-8f);
        cos_out[bs] = dot_mean / denom;
    }
}

// ---------------------------------------------------------------------------
// Kernel 3: deterministic final reduce: loss = 1 - mean(cos)
// ---------------------------------------------------------------------------
__global__ __launch_bounds__(256) void finalize_kernel(
    const float* __restrict__ cos_vals, float* __restrict__ out)
{
    __shared__ float r[256];
    const int tid = threadIdx.x;
    float acc = 0.0f;
    for (int i = tid; i < B_ * S_; i += 256) acc += cos_vals[i];
    r[tid] = acc;
    __syncthreads();
    for (int off = 128; off > 0; off >>= 1) {
        if (tid < off) r[tid] += r[tid + off];
        __syncthreads();
    }
    if (tid == 0) out[0] = 1.0f - r[0] / (float)(B_ * S_);
}

extern "C" void kernel_launch(void* const* d_in, const int* in_sizes, int n_in,
                              void* d_out, int out_size, void* d_ws, size_t ws_size,
                              hipStream_t stream) {
    const int*   attrs = (const int*)  d_in[0];   // [B, L] int32
    const float* feats = (const float*)d_in[1];   // [B, L, D] f32
    const float* Vgs   = (const float*)d_in[2];   // [B, S, D] f32
    float* out = (float*)d_out;

    float* seg  = (float*)d_ws;                          // B*S*D f32 = 8 MB
    float* cosv = seg + (size_t)B_ * S_ * D_;            // B*S f32

    seg_sum_wmma_kernel<<<B_ * 16, 128, 0, stream>>>(attrs, feats, seg);
    cos_kernel<<<B_ * S_, 256, 0, stream>>>(attrs, Vgs, seg, cosv);
    finalize_kernel<<<1, 256, 0, stream>>>(cosv, out);
}